// Conv_spe_12489764897428
// MI455X (gfx1250) — compile-verified
//
#include <hip/hip_runtime.h>

// out[m, c] = sum_k ms[m, k] * hs[c, k]
//   m in [0,8), c in [0,191), k in [0, 512*512)
// Mapped onto V_WMMA_F32_16X16X4_F32: D(16x16) = A(16x4 = ms rows) x B(4x16 = hs cols) + C
// Rows 8..15 of A and the tail channel column are *clamped duplicates* (not masked):
// their D rows/cols are never stored, so no per-load predication is needed.

typedef float v2f __attribute__((ext_vector_type(2)));
typedef float v8f __attribute__((ext_vector_type(8)));

#define BAND_MS   8
#define BAND_HS   191
#define KDIM      (512 * 512)        // 262144
#define NTILES    12                 // ceil(191 / 16)
#define KSPLIT    128                // K chunks across grid.y
#define BK        (KDIM / KSPLIT)    // 2048 k per block
#define NWAVES    8                  // 256 threads / wave32
#define KW        (BK / NWAVES)      // 256 k per wave -> 64 WMMA steps

__global__ void corr_zero_kernel(float* out, int n) {
    int i = blockIdx.x * blockDim.x + threadIdx.x;
    if (i < n) out[i] = 0.0f;
}

__global__ __launch_bounds__(256) void corr_wmma_kernel(
    const float* __restrict__ hs,   // [191, 262144]
    const float* __restrict__ ms,   // [8,   262144]
    float* __restrict__ out)        // [8, 191]
{
    __shared__ float red[NWAVES * 256];   // 8 KB: per-wave v8f accumulators

    const int tid  = threadIdx.x;
    const int wave = tid >> 5;
    const int lane = tid & 31;
    const int half = lane >> 4;           // K-pair selector within the wave
    const int lid  = lane & 15;           // M row (A) / N col (B)

    const int nbase = blockIdx.x * 16;    // first channel of this N-tile
    const int ch    = min(nbase + lid, BAND_HS - 1);  // clamp: dup col, skipped at store
    const int mrow  = lid & 7;                         // clamp: dup rows 8..15

    const float* __restrict__ hs_ptr = hs + (size_t)ch   * KDIM;
    const float* __restrict__ ms_ptr = ms + (size_t)mrow * KDIM;

    // Lane's first K index: per ISA A/B layout, lane holds K = kb + 2*half + {0,1}
    int k0 = blockIdx.y * BK + wave * KW + 2 * half;

    v8f acc = {0.0f, 0.0f, 0.0f, 0.0f, 0.0f, 0.0f, 0.0f, 0.0f};

    // 16 outer iterations x (8 unconditional b64 loads, then 4 WMMAs).
#pragma unroll 1
    for (int s = 0; s < KW / 16; ++s) {
        v2f a0 = *(const v2f*)(ms_ptr + k0);
        v2f b0 = *(const v2f*)(hs_ptr + k0);
        v2f a1 = *(const v2f*)(ms_ptr + k0 + 4);
        v2f b1 = *(const v2f*)(hs_ptr + k0 + 4);
        v2f a2 = *(const v2f*)(ms_ptr + k0 + 8);
        v2f b2 = *(const v2f*)(hs_ptr + k0 + 8);
        v2f a3 = *(const v2f*)(ms_ptr + k0 + 12);
        v2f b3 = *(const v2f*)(hs_ptr + k0 + 12);
        __builtin_prefetch(hs_ptr + k0 + 512, 0, 1);   // ~2KB ahead on the hs stream

        acc = __builtin_amdgcn_wmma_f32_16x16x4_f32(false, a0, false, b0, (short)0, acc, false, false);
        acc = __builtin_amdgcn_wmma_f32_16x16x4_f32(false, a1, false, b1, (short)0, acc, false, false);
        acc = __builtin_amdgcn_wmma_f32_16x16x4_f32(false, a2, false, b2, (short)0, acc, false, false);
        acc = __builtin_amdgcn_wmma_f32_16x16x4_f32(false, a3, false, b3, (short)0, acc, false, false);
        k0 += 16;
    }

    // Cross-wave reduction through LDS.
#pragma unroll
    for (int r = 0; r < 8; ++r)
        red[wave * 256 + r * 32 + lane] = acc[r];
    __syncthreads();

    float sum = 0.0f;
#pragma unroll
    for (int w = 0; w < NWAVES; ++w)
        sum += red[w * 256 + tid];

    // Entry tid -> (VGPR row r, lane li). D layout: VGPR r, lanes 0-15: M=r;
    // lanes 16-31: M=r+8 (duplicate rows here, filtered out below).
    const int r  = tid >> 5;
    const int li = tid & 31;
    const int m  = (li < 16) ? r : (r + 8);
    const int c  = nbase + (li & 15);
    if (m < BAND_MS && c < BAND_HS) {
        __hip_atomic_fetch_add(out + m * BAND_HS + c, sum,
                               __ATOMIC_RELAXED, __HIP_MEMORY_SCOPE_AGENT);
    }
}

extern "C" void kernel_launch(void* const* d_in, const int* in_sizes, int n_in,
                              void* d_out, int out_size, void* d_ws, size_t ws_size,
                              hipStream_t stream) {
    const float* hs = (const float*)d_in[0];   // [1, 191, 512, 512] f32
    const float* ms = (const float*)d_in[1];   // [8, 1, 512, 512]   f32
    float* out = (float*)d_out;                // [8, 191, 1, 1] -> 1528 f32

    corr_zero_kernel<<<(out_size + 255) / 256, 256, 0, stream>>>(out, out_size);

    dim3 grid(NTILES, KSPLIT, 1);
    corr_wmma_kernel<<<grid, 256, 0, stream>>>(hs, ms, out);
}